// wkv_power_45578192945854
// MI455X (gfx1250) — compile-verified
//
#include <hip/hip_runtime.h>

// ---------------------------------------------------------------------------
// RWKV WKV kernel for MI455X (gfx1250), chunked-scan formulation.
//
//   kxr[i] = sum_{j<=i} w^{i+1-j} kx[j]   (w = exp(decay_d), per channel)
//
// Chunked:  within a 16-step chunk,
//   kxr_local[i] = w^{i+1} * (L @ (w^{-j} kx[j]))[i]
// where L is the 16x16 lower-triangular ones matrix (channel independent!)
// -> computed as a 16x16x16 f32 GEMM via 4x V_WMMA_F32_16X16X4_F32.
// Cross-chunk carries are scanned by a tiny kernel, then a fully parallel
// elementwise kernel finalizes the output.
//
// Traffic ~5 MB total (multi-pass) -> sub-microsecond at 23.3 TB/s; the win
// over a naive 1024-step serial scan is parallelism (16k-way vs 256-way).
// ---------------------------------------------------------------------------

typedef __attribute__((ext_vector_type(2))) float v2f;
typedef __attribute__((ext_vector_type(8))) float v8f;

#define T_LEN   1024
#define DIMS    256
#define CHUNK   16
#define NCHUNK  (T_LEN / CHUNK)   // 64
#define DGROUPS (DIMS / 16)       // 16

// ---------------------------------------------------------------------------
// Kernel A: one wave32 per (chunk c, 16-channel group dg).
// Computes lk[i,d] = w^{i+1} * sum_{j<=i} w^{-j} kx[t0+j]  (and lv for vx),
// stores them plus the end-of-chunk carries Ak/Av = lk/lv at i=15.
// The prefix sum over j is done with WMMA f32 16x16x4 (A = triangular ones).
// ---------------------------------------------------------------------------
__global__ __launch_bounds__(32)
void wkv_chunk_wmma(const float* __restrict__ K,
                    const float* __restrict__ V,
                    const float* __restrict__ decay,
                    float* __restrict__ LK, float* __restrict__ LV,
                    float* __restrict__ Ak, float* __restrict__ Av) {
    const int c    = blockIdx.x;          // chunk index (0..63)
    const int dg   = blockIdx.y;          // channel group (0..15)
    const int lane = threadIdx.x;         // 0..31
    const int n    = lane & 15;           // N / M index inside the tile
    const int hi   = lane >> 4;           // half-wave selector
    const int d    = dg * 16 + n;         // channel
    const int t0   = c * CHUNK;           // first timestep of this chunk

    const float dec = decay[d];

    v8f yk = {0.f, 0.f, 0.f, 0.f, 0.f, 0.f, 0.f, 0.f};
    v8f yv = {0.f, 0.f, 0.f, 0.f, 0.f, 0.f, 0.f, 0.f};

    const int koff = hi * 2;   // lanes 0-15 carry K{0,1}; lanes 16-31 carry K{2,3}
    const int m    = n;        // A-matrix row held by this lane

    // 4 chained K=4 WMMAs cover K=0..15 for both the kx and vx matrices.
    #pragma unroll
    for (int kb = 0; kb < 4; ++kb) {
        const int j0 = 4 * kb + koff;
        const int j1 = j0 + 1;

        const float k0 = K[(t0 + j0) * DIMS + d];
        const float k1 = K[(t0 + j1) * DIMS + d];
        const float v0 = V[(t0 + j0) * DIMS + d];
        const float v1 = V[(t0 + j1) * DIMS + d];

        // x[j] = w^{-j} * exp(k) = exp(k - dec*j)   (fused; <= ~1e29, f32-safe)
        const float e0 = expf(k0 - dec * (float)j0);
        const float e1 = expf(k1 - dec * (float)j1);

        v2f bk; bk.x = e0;      bk.y = e1;        // B: K rows j0/j1, N = n
        v2f bv; bv.x = v0 * e0; bv.y = v1 * e1;

        // A = lower-triangular ones: L[m, j] = (j <= m)
        v2f a;  a.x = (j0 <= m) ? 1.0f : 0.0f;
                a.y = (j1 <= m) ? 1.0f : 0.0f;

        yk = __builtin_amdgcn_wmma_f32_16x16x4_f32(false, a, false, bk,
                                                   (short)0, yk, false, false);
        yv = __builtin_amdgcn_wmma_f32_16x16x4_f32(false, a, false, bv,
                                                   (short)0, yv, false, false);
    }

    // D layout: VGPR r holds row (r + 8*hi), N = n. Rescale by w^{i+1}, store.
    float lk15 = 0.f, lv15 = 0.f;
    #pragma unroll
    for (int r = 0; r < 8; ++r) {
        const int   row = r + 8 * hi;
        const float pw  = expf(dec * (float)(row + 1));   // w^{i+1}
        const float lk  = pw * yk[r];
        const float lv  = pw * yv[r];
        LK[(t0 + row) * DIMS + d] = lk;
        LV[(t0 + row) * DIMS + d] = lv;
        if (r == 7) { lk15 = lk; lv15 = lv; }
    }
    // End-of-chunk carry lives at row 15 (upper half-wave, r == 7).
    if (hi) {
        Ak[c * DIMS + d] = lk15;
        Av[c * DIMS + d] = lv15;
    }
}

// ---------------------------------------------------------------------------
// Kernel B: scan carries across the 64 chunks for each channel.
// Sk[c,d] = decayed history entering chunk c:  S_{c+1} = w^16 * S_c + A_c.
// 256 threads, 64 serial (tiny) iterations; all loads coalesced over d.
// ---------------------------------------------------------------------------
__global__ __launch_bounds__(DIMS)
void wkv_scan(const float* __restrict__ decay,
              const float* __restrict__ Ak, const float* __restrict__ Av,
              float* __restrict__ Sk, float* __restrict__ Sv) {
    const int d = threadIdx.x;
    const float wp = expf(decay[d] * (float)CHUNK);   // w^16 (>= ~3e-29, no overflow)
    float sk = 0.f, sv = 0.f;
    for (int c = 0; c < NCHUNK; ++c) {
        Sk[c * DIMS + d] = sk;
        Sv[c * DIMS + d] = sv;
        sk = wp * sk + Ak[c * DIMS + d];
        sv = wp * sv + Av[c * DIMS + d];
    }
}

// ---------------------------------------------------------------------------
// Kernel C: fully parallel elementwise finalize.
//   kxr = lk + w^{i+1} * Sk[c,d];  vxr = lv + w^{i+1} * Sv[c,d]
//   out = (vx*tf + kxr) / (kx*tf + vxr)     (crossed, faithful to reference)
// ---------------------------------------------------------------------------
__global__ __launch_bounds__(DIMS)
void wkv_finalize(const float* __restrict__ K, const float* __restrict__ V,
                  const float* __restrict__ tfirst, const float* __restrict__ decay,
                  const float* __restrict__ LK, const float* __restrict__ LV,
                  const float* __restrict__ Sk, const float* __restrict__ Sv,
                  float* __restrict__ out) {
    const int d = threadIdx.x;    // 0..255 (coalesced)
    const int t = blockIdx.x;     // 0..1023
    const int c = t >> 4;
    const int i = t & 15;

    const float dec = decay[d];
    const float kv  = K[t * DIMS + d];
    const float vv  = V[t * DIMS + d];
    const float kx  = expf(kv);
    const float vx  = vv * kx;
    const float wi  = expf(dec * (float)(i + 1));       // w^{i+1}

    const float kxr = LK[t * DIMS + d] + wi * Sk[c * DIMS + d];
    const float vxr = LV[t * DIMS + d] + wi * Sv[c * DIMS + d];
    const float tf  = expf(tfirst[d]);

    out[t * DIMS + d] = (vx * tf + kxr) / (kx * tf + vxr);
}

// ---------------------------------------------------------------------------
// Launch: inputs are k[T,D], v[T,D], time_first[D], time_decay[D] (all f32).
// Workspace usage: 4 * (64*256) + 2 * (1024*256) floats = 2.25 MB of d_ws.
// ---------------------------------------------------------------------------
extern "C" void kernel_launch(void* const* d_in, const int* in_sizes, int n_in,
                              void* d_out, int out_size, void* d_ws, size_t ws_size,
                              hipStream_t stream) {
    (void)in_sizes; (void)n_in; (void)out_size; (void)ws_size;

    const float* K      = (const float*)d_in[0];
    const float* V      = (const float*)d_in[1];
    const float* tfirst = (const float*)d_in[2];
    const float* decay  = (const float*)d_in[3];
    float*       out    = (float*)d_out;

    float* ws = (float*)d_ws;
    float* Ak = ws;                          // [NCHUNK, DIMS]
    float* Av = Ak + NCHUNK * DIMS;
    float* Sk = Av + NCHUNK * DIMS;
    float* Sv = Sk + NCHUNK * DIMS;
    float* LK = Sv + NCHUNK * DIMS;          // [T, DIMS]
    float* LV = LK + T_LEN * DIMS;

    dim3 gridA(NCHUNK, DGROUPS);             // 64 x 16 waves (one per tile)
    wkv_chunk_wmma<<<gridA, 32, 0, stream>>>(K, V, decay, LK, LV, Ak, Av);
    wkv_scan<<<1, DIMS, 0, stream>>>(decay, Ak, Av, Sk, Sv);
    wkv_finalize<<<T_LEN, DIMS, 0, stream>>>(K, V, tfirst, decay, LK, LV, Sk, Sv, out);
}